// NoiseFilter_62904091018192
// MI455X (gfx1250) — compile-verified
//
#include <hip/hip_runtime.h>
#include <stdint.h>

#define THR    0.1f
#define TPB    256
#define ITEMS  16
#define EPB    (TPB * ITEMS)   // 4096 elements per block
#define MAXSEG 32

// ---------------------------------------------------------------------------
// gfx1250 async LDS<->global DMA helpers (ASYNCcnt path, CDNA5 ISA)
// ---------------------------------------------------------------------------
__device__ __forceinline__ void async_load16_to_lds(unsigned lds, uint64_t ga) {
    // 16 bytes global -> LDS, per-lane, tracked by ASYNCcnt
    asm volatile("global_load_async_to_lds_b128 %0, %1, off"
                 :: "v"(lds), "v"(ga) : "memory");
}
__device__ __forceinline__ void async_store16_from_lds_b32x4(uint64_t ga, unsigned lds) {
    // destination is only 4B aligned (gathered region starts at odd float4
    // phase inside d_out), so issue 4 dword async stores; the immediate
    // offset applies to BOTH the LDS and global address (ISA 08 S4.4).
    asm volatile("global_store_async_from_lds_b32 %0, %1, off\n\t"
                 "global_store_async_from_lds_b32 %0, %1, off offset:4\n\t"
                 "global_store_async_from_lds_b32 %0, %1, off offset:8\n\t"
                 "global_store_async_from_lds_b32 %0, %1, off offset:12"
                 :: "v"(ga), "v"(lds) : "memory");
}
__device__ __forceinline__ void wait_async0() {
    asm volatile("s_wait_asynccnt 0" ::: "memory");
}

__device__ __forceinline__ int find_seg(int i, const int* rs, int B) {
    int seg = 0;
    while (seg + 1 < B && i >= rs[seg + 1]) seg++;
    return seg;
}

// ---------------------------------------------------------------------------
// K0: init workspace scalars
// ---------------------------------------------------------------------------
__global__ void k_init(unsigned* seg_min_bits, int* first_min, int* seg_counts,
                       int B, int N) {
    int t = threadIdx.x;
    if (t < B) {
        seg_min_bits[t] = 0x7F800000u;  // +inf bits; scores are >= 0
        first_min[t]    = N;            // sentinel (matches reference 'n')
        seg_counts[t]   = 0;
    }
}

// ---------------------------------------------------------------------------
// K1: per-segment min (uint atomicMin == float min for non-negative floats)
// ---------------------------------------------------------------------------
__global__ void k_segmin(const float* __restrict__ score,
                         const int* __restrict__ row_splits,
                         unsigned* seg_min_bits, int N, int B) {
    __shared__ int s_rs[MAXSEG + 1];
    if (threadIdx.x <= (unsigned)B) s_rs[threadIdx.x] = row_splits[threadIdx.x];
    __syncthreads();
    long stride = (long)gridDim.x * blockDim.x;
    for (long i = (long)blockIdx.x * blockDim.x + threadIdx.x; i < N; i += stride) {
        unsigned sb = __float_as_uint(score[i]);
        int seg = find_seg((int)i, s_rs, B);
        atomicMin(&seg_min_bits[seg], sb);
    }
}

// ---------------------------------------------------------------------------
// K2: per-block keep counts, per-segment keep counts, first-min global index
// ---------------------------------------------------------------------------
__global__ void k_count(const float* __restrict__ score,
                        const int* __restrict__ row_splits,
                        const unsigned* __restrict__ seg_min_bits,
                        int* first_min, int* seg_counts, int* block_counts,
                        int N, int B) {
    __shared__ int s_rs[MAXSEG + 1];
    __shared__ unsigned s_min[MAXSEG];
    __shared__ int s_cnt[MAXSEG];
    __shared__ int s_blk;
    int tid = threadIdx.x;
    if (tid <= B) s_rs[tid] = row_splits[tid];
    if (tid < B) { s_min[tid] = seg_min_bits[tid]; s_cnt[tid] = 0; }
    if (tid == 0) s_blk = 0;
    __syncthreads();

    int base = blockIdx.x * EPB + tid * ITEMS;
    int my = 0;
    for (int j = 0; j < ITEMS; j++) {
        int i = base + j;
        if (i < N) {
            unsigned sb = __float_as_uint(score[i]);
            float s = __uint_as_float(sb);
            int seg = find_seg(i, s_rs, B);
            bool ismin = (sb == s_min[seg]);
            if (ismin) atomicMin(&first_min[seg], i);
            if ((s > THR) || ismin) { my++; atomicAdd(&s_cnt[seg], 1); }
        }
    }
    atomicAdd(&s_blk, my);
    __syncthreads();
    if (tid == 0) block_counts[blockIdx.x] = s_blk;
    if (tid < B && s_cnt[tid] != 0) atomicAdd(&seg_counts[tid], s_cnt[tid]);
}

// ---------------------------------------------------------------------------
// K3: serial scans (245 + 10 values -> trivial). Writes newrs as float.
// ---------------------------------------------------------------------------
__global__ void k_scan(const int* __restrict__ block_counts, int* block_offsets,
                       const int* __restrict__ seg_counts, float* newrs_out,
                       int nblocks, int B) {
    if (threadIdx.x == 0 && blockIdx.x == 0) {
        int acc = 0;
        for (int b = 0; b < nblocks; b++) { block_offsets[b] = acc; acc += block_counts[b]; }
        int a2 = 0;
        newrs_out[0] = 0.0f;
        for (int s = 0; s < B; s++) { a2 += seg_counts[s]; newrs_out[s + 1] = (float)a2; }
    }
}

// ---------------------------------------------------------------------------
// K4: compaction pass — block exclusive scan, back[] for kept, sel[], rep_comp
// ---------------------------------------------------------------------------
__global__ void k_main(const float* __restrict__ score,
                       const int* __restrict__ row_splits,
                       const float* __restrict__ feats,
                       const unsigned* __restrict__ seg_min_bits,
                       const int* __restrict__ first_min,
                       const int* __restrict__ block_offsets,
                       float* back_out, float* gathered, int* sel, int* rep_comp,
                       int N, int B, int F, int do_inline) {
    __shared__ int s_rs[MAXSEG + 1];
    __shared__ unsigned s_min[MAXSEG];
    __shared__ int s_fm[MAXSEG];
    __shared__ int ts[TPB];
    int tid = threadIdx.x;
    if (tid <= B) s_rs[tid] = row_splits[tid];
    if (tid < B) { s_min[tid] = seg_min_bits[tid]; s_fm[tid] = first_min[tid]; }
    __syncthreads();

    int base = blockIdx.x * EPB + tid * ITEMS;
    unsigned mask = 0;
    int my = 0;
    for (int j = 0; j < ITEMS; j++) {
        int i = base + j;
        if (i < N) {
            unsigned sb = __float_as_uint(score[i]);
            float s = __uint_as_float(sb);
            int seg = find_seg(i, s_rs, B);
            if ((s > THR) || (sb == s_min[seg])) { mask |= (1u << j); my++; }
        }
    }
    ts[tid] = my;
    __syncthreads();
    // Hillis-Steele inclusive scan over 256 per-thread totals (8 steps)
    for (int off = 1; off < TPB; off <<= 1) {
        int add = (tid >= off) ? ts[tid - off] : 0;
        __syncthreads();
        ts[tid] += add;
        __syncthreads();
    }
    int comp = block_offsets[blockIdx.x] + ts[tid] - my;  // exclusive base
    for (int j = 0; j < ITEMS; j++) {
        if (mask & (1u << j)) {
            int i = base + j;
            back_out[i] = (float)comp;
            int seg = find_seg(i, s_rs, B);
            if (i == s_fm[seg]) rep_comp[seg] = comp;  // representative of segment
            if (do_inline) {
                // fallback gather (only if ws cannot hold sel[]); out region is
                // 4B aligned only -> scalar stores, vector loads
                const float4* src = (const float4*)(feats + (size_t)i * F);
                float* dst = gathered + (size_t)comp * F;
                for (int k = 0; k < F / 4; k++) {
                    float4 v = src[k];
                    dst[4 * k + 0] = v.x; dst[4 * k + 1] = v.y;
                    dst[4 * k + 2] = v.z; dst[4 * k + 3] = v.w;
                }
            } else {
                sel[comp] = i;
            }
            comp++;
        }
    }
}

// ---------------------------------------------------------------------------
// K5: fill back[] for dropped hits from segment representative
// ---------------------------------------------------------------------------
__global__ void k_fill(const float* __restrict__ score,
                       const int* __restrict__ row_splits,
                       const unsigned* __restrict__ seg_min_bits,
                       const int* __restrict__ rep_comp,
                       float* back_out, int N, int B) {
    __shared__ int s_rs[MAXSEG + 1];
    __shared__ unsigned s_min[MAXSEG];
    __shared__ int s_rep[MAXSEG];
    int tid = threadIdx.x;
    if (tid <= B) s_rs[tid] = row_splits[tid];
    if (tid < B) { s_min[tid] = seg_min_bits[tid]; s_rep[tid] = rep_comp[tid]; }
    __syncthreads();
    long stride = (long)gridDim.x * blockDim.x;
    for (long i = (long)blockIdx.x * blockDim.x + tid; i < N; i += stride) {
        unsigned sb = __float_as_uint(score[i]);
        float s = __uint_as_float(sb);
        int seg = find_seg((int)i, s_rs, B);
        bool keep = (s > THR) || (sb == s_min[seg]);
        if (!keep) back_out[i] = (float)s_rep[seg];
    }
}

// ---------------------------------------------------------------------------
// K6: gather via async LDS DMA. Each lane owns 16B chunks; double-buffered:
// overlap store(batch i) with load(batch i+1), fenced by s_wait_asynccnt.
// Data never touches VGPRs -> ASYNCcnt DMA path does the 460 MB stream.
// ---------------------------------------------------------------------------
__global__ void k_gather_async(const float* __restrict__ feats,
                               const int* __restrict__ sel,
                               float* __restrict__ outg,
                               long chunks, int F) {
    __shared__ __align__(16) unsigned char stage[2 * TPB * 16];
    int tid = threadIdx.x;
    // low 32 bits of the flat pointer == workgroup-relative LDS byte offset
    unsigned ldsA = (unsigned)(uintptr_t)(void*)(stage + tid * 16);
    unsigned ldsB = ldsA + TPB * 16;
    long stride = (long)gridDim.x * blockDim.x;
    long c = (long)blockIdx.x * blockDim.x + tid;
    const int cpr = F / 4;                    // 16B chunks per row
    const uint64_t fbase = (uint64_t)(uintptr_t)feats;
    const uint64_t obase = (uint64_t)(uintptr_t)outg;

    if (c < chunks) {
        long row = c / cpr; int q = (int)(c - row * cpr);
        async_load16_to_lds(ldsA, fbase + (uint64_t)(long)sel[row] * (F * 4) + (uint64_t)q * 16);
        while (true) {
            long n = c + stride;
            wait_async0();                               // my chunk is in LDS
            async_store16_from_lds_b32x4(obase + (uint64_t)c * 16, ldsA);
            if (n < chunks) {
                long row2 = n / cpr; int q2 = (int)(n - row2 * cpr);
                async_load16_to_lds(ldsB, fbase + (uint64_t)(long)sel[row2] * (F * 4) + (uint64_t)q2 * 16);
            }
            unsigned t = ldsA; ldsA = ldsB; ldsB = t;
            c = n;
            if (c >= chunks) break;
        }
        wait_async0();                                   // drain final stores
    }
}

// ---------------------------------------------------------------------------
extern "C" void kernel_launch(void* const* d_in, const int* in_sizes, int n_in,
                              void* d_out, int out_size, void* d_ws, size_t ws_size,
                              hipStream_t stream) {
    const float* score      = (const float*)d_in[0];
    const int*   row_splits = (const int*)d_in[1];
    const float* feats      = (const float*)d_in[2];
    // d_in[3] = tidxs: unused in forward path (loss disabled)

    const int N = in_sizes[0];
    const int B = in_sizes[1] - 1;
    const int F = in_sizes[2] / N;
    const int K = (out_size - (B + 1) - N) / F;   // kept rows (fixed by harness)

    float* newrs_out = (float*)d_out;
    float* back_out  = newrs_out + (B + 1);
    float* gathered  = back_out + N;

    const int nblocks = (N + EPB - 1) / EPB;

    int* ws = (int*)d_ws;
    unsigned* seg_min   = (unsigned*)ws;          // [0, 32)
    int* first_min      = ws + 1 * MAXSEG;        // [32, 64)
    int* seg_counts     = ws + 2 * MAXSEG;        // [64, 96)
    int* rep_comp       = ws + 3 * MAXSEG;        // [96, 128)
    int* block_counts   = ws + 4 * MAXSEG;        // [128, 128+nblocks)
    int* block_offsets  = block_counts + nblocks;
    int* sel            = ws + 8192;              // needs 128 + 2*nblocks <= 8192

    size_t need = (8192 + (size_t)K) * sizeof(int);
    int do_inline = (ws_size < need) ? 1 : 0;

    k_init<<<1, TPB, 0, stream>>>(seg_min, first_min, seg_counts, B, N);
    k_segmin<<<nblocks, TPB, 0, stream>>>(score, row_splits, seg_min, N, B);
    k_count<<<nblocks, TPB, 0, stream>>>(score, row_splits, seg_min, first_min,
                                         seg_counts, block_counts, N, B);
    k_scan<<<1, 1, 0, stream>>>(block_counts, block_offsets, seg_counts,
                                newrs_out, nblocks, B);
    k_main<<<nblocks, TPB, 0, stream>>>(score, row_splits, feats, seg_min, first_min,
                                        block_offsets, back_out, gathered, sel,
                                        rep_comp, N, B, F, do_inline);
    {
        int fb = (N + TPB - 1) / TPB; if (fb > 2048) fb = 2048;
        k_fill<<<fb, TPB, 0, stream>>>(score, row_splits, seg_min, rep_comp,
                                       back_out, N, B);
    }
    if (!do_inline) {
        long chunks = (long)K * (F / 4);
        long gb = (chunks + TPB - 1) / TPB; if (gb > 4096) gb = 4096;
        if (gb < 1) gb = 1;
        k_gather_async<<<(int)gb, TPB, 0, stream>>>(feats, sel, gathered, chunks, F);
    }
}